// FoveatedSamplingConv2d_29575144800712
// MI455X (gfx1250) — compile-verified
//
#include <hip/hip_runtime.h>
#include <hip/hip_bf16.h>

typedef __attribute__((ext_vector_type(2))) float v2f;
typedef __attribute__((ext_vector_type(8))) float v8f;

#define KK_RAW   363          // IN_C * 11 * 11
#define KK_PAD   364          // padded to multiple of 4
#define TAPS     121          // 11*11
#define H_DIM    256
#define W_DIM    256
#define HW       65536        // 256*256
#define IN_C     3
#define OUT_C    64
#define BATCH    4
#define PX_TILE  16           // pixels (w) per workgroup

// LDS budget: 93184 + 93184 + 7744 = 194112 B  (< 320 KB / WGP)
__global__ __launch_bounds__(256, 1)
void foveated_conv_wmma_f32(const float* __restrict__ x,
                            const float* __restrict__ weight,
                            const float* __restrict__ bias,
                            const int* __restrict__ sy,
                            const int* __restrict__ sx,
                            float* __restrict__ out) {
    __shared__ float lds_W[OUT_C * KK_PAD];            // [oc][kk], pad col = 0
    __shared__ float lds_S[BATCH * PX_TILE * KK_PAD];  // [b][px][kk], pad col = 0
    __shared__ int   lds_off[TAPS * PX_TILE];          // yy*256+xx per (tap, px)

    const int tid = threadIdx.x;            // 0..255
    const int h   = blockIdx.y;             // 0..255
    const int w0  = blockIdx.x * PX_TILE;   // 0,16,...,240

    // ---- Stage 1: foveated coords -> flat gather offsets (shared by b and c)
    for (int i = tid; i < TAPS * PX_TILE; i += 256) {
        int t  = i >> 4;                    // tap 0..120
        int px = i & 15;
        int cbase = (h * W_DIM + (w0 + px)) * TAPS + t;
        int yy = sy[cbase];
        int xx = sx[cbase];
        lds_off[t * PX_TILE + px] = yy * W_DIM + xx;
    }
    // ---- Stage 2: weights into LDS, padded [64][364]
    for (int i = tid; i < OUT_C * KK_PAD; i += 256) {
        int o  = i / KK_PAD;
        int kk = i - o * KK_PAD;
        lds_W[i] = (kk < KK_RAW) ? weight[o * KK_RAW + kk] : 0.0f;
    }
    __syncthreads();

    // ---- Stage 3: gather S[b][px][kk] = x[b, c, sy, sx]
    for (int i = tid; i < BATCH * PX_TILE * KK_PAD; i += 256) {
        int kk   = i % KK_PAD;
        int rest = i / KK_PAD;
        int px   = rest & 15;
        int b    = rest >> 4;
        float v = 0.0f;
        if (kk < KK_RAW) {
            int c = kk / TAPS;
            int t = kk - c * TAPS;
            v = x[(b * IN_C + c) * HW + lds_off[t * PX_TILE + px]];
        }
        lds_S[i] = v;
    }
    __syncthreads();

    // ---- Stage 4: GEMM via V_WMMA_F32_16X16X4_F32
    // wave -> (batch, oc half); each wave owns two 16(oc) x 16(px) tiles.
    const int wave  = tid >> 5;             // 0..7
    const int lane  = tid & 31;
    const int b     = wave & 3;
    const int mhalf = wave >> 2;            // 0 or 1  -> oc base 0 / 32
    const int lm    = lane & 15;
    const int koff  = (lane >> 4) << 1;     // 0 (lanes 0-15) or 2 (lanes 16-31)

    const float* Sb = lds_S + (b * PX_TILE + lm) * KK_PAD;        // B: row = px(N)
    const float* W0 = lds_W + (mhalf * 32 + lm) * KK_PAD;         // A tile 0: row = oc(M)
    const float* W1 = W0 + 16 * KK_PAD;                           // A tile 1

    v8f acc0 = {};
    v8f acc1 = {};
    for (int k0 = 0; k0 < KK_PAD; k0 += 4) {
        int kk = k0 + koff;
        v2f a0 = { W0[kk], W0[kk + 1] };    // A 16x4: M=lm, K={kk,kk+1}
        v2f a1 = { W1[kk], W1[kk + 1] };
        v2f bv = { Sb[kk], Sb[kk + 1] };    // B 4x16: N=lm, K={kk,kk+1}
        acc0 = __builtin_amdgcn_wmma_f32_16x16x4_f32(
            false, a0, false, bv, (short)0, acc0, false, false);
        acc1 = __builtin_amdgcn_wmma_f32_16x16x4_f32(
            false, a1, false, bv, (short)0, acc1, false, false);
    }

    // ---- Stage 5: epilogue — C/D layout: VGPR r holds M=r (lanes 0-15) and
    // M=r+8 (lanes 16-31), N = lane&15.
    const int n    = lane & 15;
    const int mtop = (lane >> 4) << 3;      // 0 or 8
    const int wcol = w0 + n;
    const int hw_off = h * W_DIM + wcol;

    #pragma unroll
    for (int r = 0; r < 8; ++r) {
        int oc0 = mhalf * 32 + r + mtop;          // tile 0
        out[(b * OUT_C + oc0) * HW + hw_off] = acc0[r] + bias[oc0];
        int oc1 = oc0 + 16;                        // tile 1
        out[(b * OUT_C + oc1) * HW + hw_off] = acc1[r] + bias[oc1];
    }
}

extern "C" void kernel_launch(void* const* d_in, const int* in_sizes, int n_in,
                              void* d_out, int out_size, void* d_ws, size_t ws_size,
                              hipStream_t stream) {
    const float* x      = (const float*)d_in[0];
    const float* weight = (const float*)d_in[1];
    const float* bias   = (const float*)d_in[2];
    const int*   sy     = (const int*)d_in[3];
    const int*   sx     = (const int*)d_in[4];
    float* out = (float*)d_out;

    dim3 grid(W_DIM / PX_TILE, H_DIM);   // 16 x 256 = 4096 workgroups
    dim3 block(256);
    foveated_conv_wmma_f32<<<grid, block, 0, stream>>>(x, weight, bias, sy, sx, out);
}